// PDP_36532991820366
// MI455X (gfx1250) — compile-verified
//
#include <hip/hip_runtime.h>
#include <cstdint>

typedef unsigned int u32x4 __attribute__((ext_vector_type(4)));
typedef int          i32x8 __attribute__((ext_vector_type(8)));
typedef int          i32x4 __attribute__((ext_vector_type(4)));

#define TILE 2048
#define BLK  256

// ---- workspace layout (uint32 units) ----
#define OFF_H1    0u      // 4096 bins  (level-1: top 12 bits of |w| bits)
#define OFF_H2    4096u   // 2 x 4096   (level-2, per rank target)
#define OFF_H3    12288u  // 2 x 256    (level-3, per rank target)
#define OFF_STATE 12800u  // [0..1]=remaining asc rank, [2..3]=bit-prefix, [4]=t^2 bits

__device__ __forceinline__ uint32_t lds_off_of(const void* p) {
  // flat LDS address: low 32 bits are the workgroup-relative LDS byte offset
  return (uint32_t)(uintptr_t)p;
}

// Issue one TDM transfer: nelem contiguous f32 from global -> LDS offset.
// D# per CDNA5 ISA ch.8: group0 {count=1, lds_addr, global_addr, type=2},
// group1 {data_size=4B, tensor_dim0=nelem, tensor_dim1=1, tile_dim0=nelem}.
__device__ __forceinline__ void tdm_load(uint32_t lds_off, const float* g, uint32_t nelem) {
  uint64_t ga = (uint64_t)(uintptr_t)g;
  u32x4 g0;
  g0.x = 1u;                                                   // count=1, user D#
  g0.y = lds_off;                                              // lds_addr (bytes)
  g0.z = (uint32_t)ga;                                         // global_addr[31:0]
  g0.w = (uint32_t)((ga >> 32) & 0x01FFFFFFu) | 0x80000000u;   // addr[56:32] | type=2
  i32x8 g1;
  g1[0] = (int)(2u << 16);                                     // data_size = 4 bytes
  g1[1] = (int)((nelem & 0xFFFFu) << 16);                      // tensor_dim0[15:0]
  g1[2] = (int)(((nelem >> 16) & 0xFFFFu) | (1u << 16));       // tensor_dim0[31:16], tensor_dim1=1
  g1[3] = (int)((nelem & 0xFFFFu) << 16);                      // tile_dim0
  g1[4] = 0;                                                   // tile_dim1=0, tile_dim2=0
  g1[5] = (int)nelem;                                          // tensor_dim0_stride
  g1[6] = 0;
  g1[7] = 0;
  i32x4 g2 = {0, 0, 0, 0};
  i32x4 g3 = {0, 0, 0, 0};
#if defined(__clang_major__) && (__clang_major__ >= 23)
  i32x8 g4 = {0, 0, 0, 0, 0, 0, 0, 0};
  __builtin_amdgcn_tensor_load_to_lds(g0, g1, g2, g3, g4, 0);
#else
  __builtin_amdgcn_tensor_load_to_lds(g0, g1, g2, g3, 0);
#endif
}

// Double-buffered TDM streaming over grid-strided tiles; proc(tile, base, ne).
template <class F>
__device__ __forceinline__ void stream_tiles(const float* __restrict__ w, long long n,
                                             float (*buf)[TILE], F&& proc) {
  const int tid = (int)threadIdx.x;
  const long long ntiles = (n + TILE - 1) / TILE;
  long long t = (long long)blockIdx.x;
  const long long stride = (long long)gridDim.x;
  const bool issuer = (tid < 32);   // wave 0 issues TDM + owns TENSORcnt
  if (t >= ntiles) return;
  {
    long long base = t * TILE;
    uint32_t ne = (uint32_t)(((n - base) < (long long)TILE) ? (n - base) : (long long)TILE);
    if (issuer) tdm_load(lds_off_of(&buf[0][0]), w + base, ne);
  }
  int cur = 0;
  for (; t < ntiles; t += stride) {
    long long base = t * TILE;
    uint32_t ne = (uint32_t)(((n - base) < (long long)TILE) ? (n - base) : (long long)TILE);
    if (issuer) __builtin_amdgcn_s_wait_tensorcnt(0);   // buf[cur] ready
    __syncthreads();
    long long nt = t + stride;
    if (nt < ntiles && issuer) {                        // prefetch next tile into other buffer
      long long nb = nt * TILE;
      uint32_t nne = (uint32_t)(((n - nb) < (long long)TILE) ? (n - nb) : (long long)TILE);
      tdm_load(lds_off_of(&buf[cur ^ 1][0]), w + nb, nne);
    }
    proc(buf[cur], base, ne);
    __syncthreads();                                    // done reading buf[cur] before reuse
    cur ^= 1;
  }
}

// ---- pass 1: 4096-bin histogram of (key >> 20), key = bits(|w|) ----
__global__ void k_hist1(const float* __restrict__ w, long long n, uint32_t* __restrict__ ws) {
  __shared__ float buf[2][TILE];
  __shared__ uint32_t h[2][4096];          // wave-replicated to cut LDS atomic conflicts
  const int tid = (int)threadIdx.x;
  for (int b = tid; b < 4096; b += BLK) { h[0][b] = 0u; h[1][b] = 0u; }
  __syncthreads();
  uint32_t* myh = h[(tid >> 5) & 1];
  stream_tiles(w, n, buf, [&](float* tile, long long, uint32_t ne) {
    for (int j = tid; j < (int)ne; j += BLK) {
      uint32_t key = __float_as_uint(tile[j]) & 0x7fffffffu;
      atomicAdd(&myh[key >> 20], 1u);
    }
  });
  __syncthreads();
  for (int b = tid; b < 4096; b += BLK) {
    uint32_t s = h[0][b] + h[1][b];
    if (s) atomicAdd(&ws[OFF_H1 + b], s);
  }
}

// ---- passes 2/3: histogram of next digit for elements matching each target's prefix ----
__global__ void k_hist_match(const float* __restrict__ w, long long n, uint32_t* __restrict__ ws,
                             int matchShift, int digitShift, uint32_t digitMask,
                             uint32_t histOff, uint32_t histStride, uint32_t nbins) {
  __shared__ float buf[2][TILE];
  __shared__ uint32_t h[2][4096];          // h[j] = histogram for rank target j
  const int tid = (int)threadIdx.x;
  const uint32_t sel0 = ws[OFF_STATE + 2];
  const uint32_t sel1 = ws[OFF_STATE + 3];
  for (int b = tid; b < 4096; b += BLK) { h[0][b] = 0u; h[1][b] = 0u; }
  __syncthreads();
  stream_tiles(w, n, buf, [&](float* tile, long long, uint32_t ne) {
    for (int j = tid; j < (int)ne; j += BLK) {
      uint32_t key  = __float_as_uint(tile[j]) & 0x7fffffffu;
      uint32_t pref = key >> matchShift;
      uint32_t d    = (key >> digitShift) & digitMask;
      if (pref == sel0) atomicAdd(&h[0][d], 1u);
      if (pref == sel1) atomicAdd(&h[1][d], 1u);
    }
  });
  __syncthreads();
  for (int b = tid; b < (int)nbins; b += BLK) {
    uint32_t s0 = h[0][b];
    if (s0) atomicAdd(&ws[histOff + b], s0);
    uint32_t s1 = h[1][b];
    if (s1) atomicAdd(&ws[histOff + histStride + b], s1);
  }
}

// ---- scan: locate the bin holding each target rank; refine prefix & remaining rank ----
__global__ void k_scan(uint32_t* __restrict__ ws, uint32_t histOff, uint32_t histStride,
                       uint32_t nbins, int selShift, int finalLevel) {
  __shared__ uint32_t part[BLK];
  const int tid = (int)threadIdx.x;
  const uint32_t seg = nbins / BLK;        // 16 (4096 bins) or 1 (256 bins)
  for (int j = 0; j < 2; ++j) {
    const uint32_t* h = ws + histOff + (uint32_t)j * histStride;
    const uint32_t target = ws[OFF_STATE + j];
    uint32_t local = 0;
    for (uint32_t k = 0; k < seg; ++k) local += h[(uint32_t)tid * seg + k];
    part[tid] = local;
    __syncthreads();
    if (tid == 0) {                        // exclusive prefix over 256 partials
      uint32_t c = 0;
      for (int i = 0; i < BLK; ++i) { uint32_t v = part[i]; part[i] = c; c += v; }
    }
    __syncthreads();
    const uint32_t pre = part[tid];
    if (target >= pre && target < pre + local) {   // exactly one thread
      uint32_t c = pre;
      for (uint32_t k = 0; k < seg; ++k) {
        uint32_t hv = h[(uint32_t)tid * seg + k];
        if (target < c + hv) {
          uint32_t bin = (uint32_t)tid * seg + k;
          ws[OFF_STATE + j] = target - c;                              // rank within bin
          ws[OFF_STATE + 2 + j] = (ws[OFF_STATE + 2 + j] << selShift) | bin;
          break;
        }
        c += hv;
      }
    }
    __syncthreads();
  }
  if (finalLevel && tid == 0) {
    float v0 = __uint_as_float(ws[OFF_STATE + 2]);   // desc[lim]
    float v1 = __uint_as_float(ws[OFF_STATE + 3]);   // desc[lim+1]
    float t = 0.5f * (v0 + v1);
    ws[OFF_STATE + 4] = __float_as_uint(t * t);
  }
}

// ---- final pass: out = w * sigmoid((w^2 - t^2)/TEMP) ----
__global__ void k_mask(const float* __restrict__ w, float* __restrict__ out,
                       long long n, const uint32_t* __restrict__ ws) {
  __shared__ float buf[2][TILE];
  const int tid = (int)threadIdx.x;
  const float t2 = __uint_as_float(ws[OFF_STATE + 4]);
  stream_tiles(w, n, buf, [&](float* tile, long long base, uint32_t ne) {
    if (ne == TILE) {
      const float4* b4 = (const float4*)tile;
      float4* o4 = (float4*)(out + base);
      for (int i = tid; i < TILE / 4; i += BLK) {
        float4 v = b4[i];
        float4 r;
        r.x = v.x / (1.0f + expf((t2 - v.x * v.x) * 1.0e5f));
        r.y = v.y / (1.0f + expf((t2 - v.y * v.y) * 1.0e5f));
        r.z = v.z / (1.0f + expf((t2 - v.z * v.z) * 1.0e5f));
        r.w = v.w / (1.0f + expf((t2 - v.w * v.w) * 1.0e5f));
        o4[i] = r;
      }
    } else {
      for (int j = tid; j < (int)ne; j += BLK) {
        float x = tile[j];
        out[base + j] = x / (1.0f + expf((t2 - x * x) * 1.0e5f));
      }
    }
  });
}

__global__ void k_init(uint32_t* __restrict__ ws, uint32_t a0, uint32_t a1) {
  int i = (int)(blockIdx.x * blockDim.x + threadIdx.x);
  for (int b = i; b < (int)OFF_STATE; b += (int)(gridDim.x * blockDim.x)) ws[b] = 0u;
  if (i == 0) {
    ws[OFF_STATE + 0] = a0;   // ascending rank of desc[lim]
    ws[OFF_STATE + 1] = a1;   // ascending rank of desc[lim+1]
    ws[OFF_STATE + 2] = 0u;
    ws[OFF_STATE + 3] = 0u;
    ws[OFF_STATE + 4] = 0u;
  }
}

extern "C" void kernel_launch(void* const* d_in, const int* in_sizes, int n_in,
                              void* d_out, int out_size, void* d_ws, size_t ws_size,
                              hipStream_t stream) {
  const float* w = (const float*)d_in[0];
  float* out = (float*)d_out;
  uint32_t* ws = (uint32_t*)d_ws;
  long long n = (long long)in_sizes[0];
  if (n < 2) return;

  // lim = clamp(int((1-SPARSITY)*n), 0, n-2); SPARSITY = 0.5
  long long lim = (long long)(0.5 * (double)n);
  if (lim < 0) lim = 0;
  if (lim > n - 2) lim = n - 2;
  uint32_t a0 = (uint32_t)(n - 1 - lim);   // asc rank of desc[lim]
  uint32_t a1 = (uint32_t)(n - 2 - lim);   // asc rank of desc[lim+1]

  k_init<<<32, BLK, 0, stream>>>(ws, a0, a1);

  const int grid = 1024;
  // level 1: top 12 bits
  k_hist1<<<grid, BLK, 0, stream>>>(w, n, ws);
  k_scan<<<1, BLK, 0, stream>>>(ws, OFF_H1, 0u, 4096u, 12, 0);
  // level 2: bits [19:8]
  k_hist_match<<<grid, BLK, 0, stream>>>(w, n, ws, 20, 8, 0xFFFu, OFF_H2, 4096u, 4096u);
  k_scan<<<1, BLK, 0, stream>>>(ws, OFF_H2, 4096u, 4096u, 12, 0);
  // level 3: bits [7:0]
  k_hist_match<<<grid, BLK, 0, stream>>>(w, n, ws, 8, 0, 0xFFu, OFF_H3, 256u, 256u);
  k_scan<<<1, BLK, 0, stream>>>(ws, OFF_H3, 256u, 256u, 8, 1);
  // elementwise soft mask
  k_mask<<<2048, BLK, 0, stream>>>(w, out, n, ws);
}